// Hopfield_Kuramoto_additive_75110388072807
// MI455X (gfx1250) — compile-verified
//
#include <hip/hip_runtime.h>
#include <hip/hip_bf16.h>

typedef __attribute__((ext_vector_type(2))) float v2f;
typedef __attribute__((ext_vector_type(8))) float v8f;

#define DDIM 64
#define NMAX 8192

__device__ __forceinline__ float wave_reduce_sum(float v) {
#pragma unroll
    for (int off = 16; off > 0; off >>= 1) v += __shfl_xor(v, off, 32);
    return v;
}

// ---------------------------------------------------------------------------
// Kernel 1: zero f_K accumulator, g = tanh(state_H), skew = 0.5*(omega-omega^T)
// ---------------------------------------------------------------------------
__global__ void k_prep(const float* __restrict__ sH, const float* __restrict__ omega,
                       float* __restrict__ facc, float* __restrict__ gv,
                       float* __restrict__ skew, int N) {
    int i = blockIdx.x * blockDim.x + threadIdx.x;
    int ND = N * DDIM;
    int stride = gridDim.x * blockDim.x;
    for (int t = i; t < ND; t += stride) facc[t] = 0.0f;
    if (i < N) gv[i] = tanhf(sH[i]);
    if (i < DDIM * DDIM) {
        int k = i / DDIM, n = i % DDIM;
        skew[i] = 0.5f * (omega[k * DDIM + n] - omega[n * DDIM + k]);
    }
}

// ---------------------------------------------------------------------------
// Kernel 2: row-normalize state_K (wave per row, lane covers d and d+32)
// ---------------------------------------------------------------------------
__global__ void k_norm(const float* __restrict__ K, float* __restrict__ Kn, int N) {
    int lane = threadIdx.x & 31, wid = threadIdx.x >> 5;
    int row = blockIdx.x * (blockDim.x >> 5) + wid;
    if (row >= N) return;
    float a0 = K[row * DDIM + lane];
    float a1 = K[row * DDIM + lane + 32];
    float s = wave_reduce_sum(a0 * a0 + a1 * a1);
    float inv = rsqrtf(s);
    Kn[row * DDIM + lane]      = a0 * inv;
    Kn[row * DDIM + lane + 32] = a1 * inv;
}

// ---------------------------------------------------------------------------
// Kernel 3: f_H = -state_H + W_H @ g   (HBM-bound: 256 MB of W_H)
// wave-per-row; g staged in LDS; lane does strided float4 (b128) loads
// ---------------------------------------------------------------------------
__global__ void k_matvec(const float* __restrict__ W, const float* __restrict__ gv,
                         const float* __restrict__ sH, float* __restrict__ fH, int N) {
    __shared__ float gs[NMAX];
    for (int t = threadIdx.x; t < N && t < NMAX; t += blockDim.x) gs[t] = gv[t];
    __syncthreads();
    int lane = threadIdx.x & 31, wid = threadIdx.x >> 5;
    int wpb = blockDim.x >> 5;
    int gw = blockIdx.x * wpb + wid;
    int tw = gridDim.x * wpb;
    int n4 = N >> 2;
    for (int row = gw; row < N; row += tw) {
        const float4* Wr = (const float4*)(W + (size_t)row * N);
        float acc = 0.0f;
        for (int c4 = lane; c4 < n4; c4 += 32) {
            float4 w = Wr[c4];
            int c = c4 << 2;
            acc += w.x * gs[c] + w.y * gs[c + 1] + w.z * gs[c + 2] + w.w * gs[c + 3];
        }
        acc = wave_reduce_sum(acc);
        if (lane == 0) fH[row] = -sH[row] + acc;
    }
}

// ---------------------------------------------------------------------------
// Kernel 4: HK edges. Gram = <Kn[i],Kn[j]>; f_H atomics; -g_i*g_j/2 * Kn scatters
// wave per edge; lane covers components d and d+32
// ---------------------------------------------------------------------------
__global__ void k_hk(const int* __restrict__ ind, const float* __restrict__ Kn,
                     const float* __restrict__ gv, float* __restrict__ fH,
                     float* __restrict__ facc, int E) {
    int lane = threadIdx.x & 31, wid = threadIdx.x >> 5;
    int e = blockIdx.x * (blockDim.x >> 5) + wid;
    if (e >= E) return;
    int i = ind[2 * e], j = ind[2 * e + 1];
    float a0 = Kn[i * DDIM + lane], a1 = Kn[i * DDIM + lane + 32];
    float c0 = Kn[j * DDIM + lane], c1 = Kn[j * DDIM + lane + 32];
    float Gram = wave_reduce_sum(a0 * c0 + a1 * c1);
    float gi = gv[i], gj = gv[j];
    if (lane == 0) {
        atomicAdd(&fH[i], 0.5f * Gram * gj);   // / KAPPA_H
        atomicAdd(&fH[j], 0.5f * Gram * gi);
    }
    float G = 0.5f * gi * gj;                  // / KAPPA_K folded in
    atomicAdd(&facc[i * DDIM + lane],      -G * c0);
    atomicAdd(&facc[i * DDIM + lane + 32], -G * c1);
    atomicAdd(&facc[j * DDIM + lane],      -G * a0);
    atomicAdd(&facc[j * DDIM + lane + 32], -G * a1);
}

// ---------------------------------------------------------------------------
// Kernel 5: K edges. s = <Kn[p],Kn[q]>; dE = MLP(s); scatter dE*Kn
// MLP hidden H=64 distributed as lane and lane+32, second wave reduction
// ---------------------------------------------------------------------------
__global__ void k_kedges(const int* __restrict__ ind, const float* __restrict__ Kn,
                         const float* __restrict__ w1v, const float* __restrict__ b1v,
                         const float* __restrict__ w2v, float* __restrict__ facc, int E) {
    int lane = threadIdx.x & 31, wid = threadIdx.x >> 5;
    int e = blockIdx.x * (blockDim.x >> 5) + wid;
    if (e >= E) return;
    int p = ind[2 * e], q = ind[2 * e + 1];
    float a0 = Kn[p * DDIM + lane], a1 = Kn[p * DDIM + lane + 32];
    float c0 = Kn[q * DDIM + lane], c1 = Kn[q * DDIM + lane + 32];
    float s = wave_reduce_sum(a0 * c0 + a1 * c1);
    float h0 = tanhf(s * w1v[lane]      + b1v[lane]);
    float h1 = tanhf(s * w1v[lane + 32] + b1v[lane + 32]);
    float dE = wave_reduce_sum(h0 * w2v[lane] + h1 * w2v[lane + 32]);
    atomicAdd(&facc[p * DDIM + lane],      dE * c0);
    atomicAdd(&facc[p * DDIM + lane + 32], dE * c1);
    atomicAdd(&facc[q * DDIM + lane],      dE * a0);
    atomicAdd(&facc[q * DDIM + lane + 32], dE * a1);
}

// ---------------------------------------------------------------------------
// Kernel 6: Komega = Kn @ skew   (8192x64 @ 64x64 f32 GEMM via WMMA 16x16x4)
// wave per 16-row tile; 4 v8f accumulators (cols 0-63); skew staged in LDS
// A layout: lanes 0-15 = M, VGPR0/1 = K+0/K+1; lanes 16-31 = K+2/K+3
// ---------------------------------------------------------------------------
__global__ void k_komega(const float* __restrict__ Kn, const float* __restrict__ skew,
                         float* __restrict__ Kom, int N) {
    __shared__ float ssk[DDIM * DDIM];
    for (int t = threadIdx.x; t < DDIM * DDIM; t += blockDim.x) ssk[t] = skew[t];
    __syncthreads();
    int lane = threadIdx.x & 31, wid = threadIdx.x >> 5;
    int tile = blockIdx.x * (blockDim.x >> 5) + wid;
    if (tile >= (N >> 4)) return;
    int row0 = tile << 4;
    int l15 = lane & 15;
    int hi = lane >> 4;          // 0 or 1
    int arow = row0 + l15;
    int kofs = hi * 2;           // K sub-offset for this lane half
    v8f acc[4] = {};
#pragma unroll
    for (int k0 = 0; k0 < DDIM; k0 += 4) {
        v2f A;
        A.x = Kn[arow * DDIM + k0 + kofs];
        A.y = Kn[arow * DDIM + k0 + kofs + 1];
#pragma unroll
        for (int t = 0; t < 4; ++t) {
            v2f B;
            int n = (t << 4) + l15;
            B.x = ssk[(k0 + kofs) * DDIM + n];
            B.y = ssk[(k0 + kofs + 1) * DDIM + n];
            acc[t] = __builtin_amdgcn_wmma_f32_16x16x4_f32(
                false, A, false, B, (short)0, acc[t], false, false);
        }
    }
    // C/D layout: VGPR v: lanes 0-15 -> row row0+v, lanes 16-31 -> row row0+v+8
    int crow = row0 + hi * 8;
#pragma unroll
    for (int t = 0; t < 4; ++t) {
#pragma unroll
        for (int v = 0; v < 8; ++v) {
            Kom[(crow + v) * DDIM + (t << 4) + l15] = acc[t][v];
        }
    }
}

// ---------------------------------------------------------------------------
// Kernel 7: f_K = -facc + Kn * <Kn,facc>_row + Komega   (wave per row)
// ---------------------------------------------------------------------------
__global__ void k_final(const float* __restrict__ Kn, const float* __restrict__ facc,
                        const float* __restrict__ Kom, float* __restrict__ fK, int N) {
    int lane = threadIdx.x & 31, wid = threadIdx.x >> 5;
    int row = blockIdx.x * (blockDim.x >> 5) + wid;
    if (row >= N) return;
    float k0 = Kn[row * DDIM + lane],   k1 = Kn[row * DDIM + lane + 32];
    float r0 = facc[row * DDIM + lane], r1 = facc[row * DDIM + lane + 32];
    float dot = wave_reduce_sum(k0 * r0 + k1 * r1);
    fK[row * DDIM + lane]      = -r0 + k0 * dot + Kom[row * DDIM + lane];
    fK[row * DDIM + lane + 32] = -r1 + k1 * dot + Kom[row * DDIM + lane + 32];
}

// ---------------------------------------------------------------------------
extern "C" void kernel_launch(void* const* d_in, const int* in_sizes, int n_in,
                              void* d_out, int out_size, void* d_ws, size_t ws_size,
                              hipStream_t stream) {
    const float* sH    = (const float*)d_in[0];
    const float* sK    = (const float*)d_in[1];
    const float* WH    = (const float*)d_in[2];
    const float* omega = (const float*)d_in[3];
    const float* w1v   = (const float*)d_in[4];
    const float* b1v   = (const float*)d_in[5];
    const float* w2v   = (const float*)d_in[6];
    const int*   indK  = (const int*)d_in[7];
    const int*   indHK = (const int*)d_in[8];

    int N   = in_sizes[0];
    int EK  = in_sizes[7] / 2;
    int EHK = in_sizes[8] / 2;
    int ND  = N * DDIM;

    float* fH = (float*)d_out;
    float* fK = (float*)d_out + N;

    float* Kn   = (float*)d_ws;           // N*D
    float* gv   = Kn + (size_t)ND;        // N
    float* skew = gv + N;                 // D*D
    float* facc = skew + DDIM * DDIM;     // N*D
    float* Kom  = facc + (size_t)ND;      // N*D

    k_prep  <<<(ND + 255) / 256,        256, 0, stream>>>(sH, omega, facc, gv, skew, N);
    k_norm  <<<(N + 7) / 8,             256, 0, stream>>>(sK, Kn, N);
    k_matvec<<<256,                     256, 0, stream>>>(WH, gv, sH, fH, N);
    k_hk    <<<(EHK + 7) / 8,           256, 0, stream>>>(indHK, Kn, gv, fH, facc, EHK);
    k_kedges<<<(EK + 7) / 8,            256, 0, stream>>>(indK, Kn, w1v, b1v, w2v, facc, EK);
    k_komega<<<((N >> 4) + 7) / 8,      256, 0, stream>>>(Kn, skew, Kom, N);
    k_final <<<(N + 7) / 8,             256, 0, stream>>>(Kn, facc, Kom, fK, N);
}